// WeightedCrossEntropyLoss_45543833206824
// MI455X (gfx1250) — compile-verified
//
#include <hip/hip_runtime.h>
#include <hip/hip_bf16.h>
#include <math.h>

#define BATCH 16
#define NC    3
#define H     512
#define W     512
#define KSIZE 15
#define PAD   7
#define HW    (H * W)
#define BHW   (BATCH * HW)

typedef float v2f __attribute__((ext_vector_type(2)));
typedef float v8f __attribute__((ext_vector_type(8)));

__device__ __forceinline__ int clampi(int v, int lo, int hi) {
  v = v < lo ? lo : v;
  return v > hi ? hi : v;
}

// ---------------------------------------------------------------------------
// Pass 1: horizontal 15-tap box SUM of obj = (target > 0), via WMMA.
// One wave32 per 16x16 output tile, 4 waves per block.
// D = A(16x4 data) x B(4x16 band), 8 K-chunks covering input columns
// [x0-7, x0+25). Band B[K][j] = 1 iff K-j in [0,14].
// Phase 1 gathers all 16 operand values with UNCONDITIONAL clamped loads
// (boundary zeros applied via select, not branches) so loads pipeline;
// phase 2 runs the 8 chained WMMAs.
// ---------------------------------------------------------------------------
__global__ __launch_bounds__(128)
void hbox_kernel(const int* __restrict__ target, float* __restrict__ tmp) {
  int tile = blockIdx.x * 4 + (threadIdx.x >> 5);
  int tx = tile & 31;          // W/16 tiles
  int ty = (tile >> 5) & 31;   // H/16 tiles
  int b  = tile >> 10;
  int x0 = tx << 4, y0 = ty << 4;

  int lane = threadIdx.x & 31;
  int nj = lane & 15;          // row M for A; column N for B/C/D
  int hi = lane >> 4;          // selects K-pair {0,1} vs {2,3}

  const int* tgt_row = target + (b * H + (y0 + nj)) * W;

  // ---- gather phase: 16 unconditional loads, branch-free edge handling ----
  float av[16];
  #pragma unroll
  for (int c = 0; c < 8; ++c) {
    int kbase = c * 4 + hi * 2;
    int g0 = x0 - PAD + kbase;
    int g1 = g0 + 1;
    int t0 = tgt_row[clampi(g0, 0, W - 1)];
    int t1 = tgt_row[clampi(g1, 0, W - 1)];
    av[2 * c]     = (g0 >= 0 && g0 < W && t0 > 0) ? 1.0f : 0.0f;
    av[2 * c + 1] = (g1 >= 0 && g1 < W && t1 > 0) ? 1.0f : 0.0f;
  }

  // ---- compute phase: 8 chained WMMAs against the constant band matrix ----
  v8f acc = {};
  #pragma unroll
  for (int c = 0; c < 8; ++c) {
    int kbase = c * 4 + hi * 2;
    v2f a;  a.x = av[2 * c];  a.y = av[2 * c + 1];
    int d0 = kbase - nj;
    int d1 = d0 + 1;
    v2f bm;
    bm.x = ((unsigned)d0 <= 14u) ? 1.0f : 0.0f;
    bm.y = ((unsigned)d1 <= 14u) ? 1.0f : 0.0f;
    acc = __builtin_amdgcn_wmma_f32_16x16x4_f32(false, a, false, bm,
                                                (short)0, acc, false, false);
  }

  // C/D layout: VGPR v -> row M = v (lanes 0-15) or v+8 (lanes 16-31), N = nj
  float* out_base = tmp + (b * H + y0) * W + x0;
  #pragma unroll
  for (int v = 0; v < 8; ++v)
    out_base[(v + hi * 8) * W + nj] = acc[v];
}

// ---------------------------------------------------------------------------
// Pass 2: vertical 15-tap box sum of tmp, scaled 1/225 -> density.
// Roles swap: A = constant band (value 1/225), B = data rows (clamped loads).
// ---------------------------------------------------------------------------
__global__ __launch_bounds__(128)
void vbox_kernel(const float* __restrict__ tmp, float* __restrict__ density) {
  int tile = blockIdx.x * 4 + (threadIdx.x >> 5);
  int tx = tile & 31;
  int ty = (tile >> 5) & 31;
  int b  = tile >> 10;
  int x0 = tx << 4, y0 = ty << 4;

  int lane = threadIdx.x & 31;
  int nj = lane & 15;
  int hi = lane >> 4;

  const float* src = tmp + (size_t)b * HW + x0 + nj;
  const float inv = 1.0f / (float)(KSIZE * KSIZE);

  // ---- gather phase ----
  float bv[16];
  #pragma unroll
  for (int c = 0; c < 8; ++c) {
    int kbase = c * 4 + hi * 2;
    int r0 = y0 - PAD + kbase;
    int r1 = r0 + 1;
    float t0 = src[clampi(r0, 0, H - 1) * W];
    float t1 = src[clampi(r1, 0, H - 1) * W];
    bv[2 * c]     = (r0 >= 0 && r0 < H) ? t0 : 0.0f;
    bv[2 * c + 1] = (r1 >= 0 && r1 < H) ? t1 : 0.0f;
  }

  // ---- compute phase ----
  v8f acc = {};
  #pragma unroll
  for (int c = 0; c < 8; ++c) {
    int kbase = c * 4 + hi * 2;
    int d0 = kbase - nj;
    int d1 = d0 + 1;
    v2f a;
    a.x = ((unsigned)d0 <= 14u) ? inv : 0.0f;
    a.y = ((unsigned)d1 <= 14u) ? inv : 0.0f;
    v2f bm;  bm.x = bv[2 * c];  bm.y = bv[2 * c + 1];
    acc = __builtin_amdgcn_wmma_f32_16x16x4_f32(false, a, false, bm,
                                                (short)0, acc, false, false);
  }

  float* out_base = density + (size_t)b * HW + y0 * W + x0;
  #pragma unroll
  for (int v = 0; v < 8; ++v)
    out_base[(v + hi * 8) * W + nj] = acc[v];
}

// ---------------------------------------------------------------------------
// Pass 3: position weights. One thread per (b, x) column; consecutive threads
// hit consecutive x, so each row iteration is a coalesced 2KB load.
// fy = first row with mask==1 (0 if none); w = 1 + 1.5*0.8^(fy-y) for y<fy.
// ---------------------------------------------------------------------------
__global__ __launch_bounds__(256)
void posw_kernel(const int* __restrict__ mask, float* __restrict__ posw) {
  int id = blockIdx.x * blockDim.x + threadIdx.x;   // b*W + x
  if (id >= BATCH * W) return;
  int b = id / W, x = id - b * W;
  const int* mcol = mask + (size_t)b * HW + x;
  int fy = 0;
  for (int y = 0; y < H; ++y) {
    if (mcol[y * W] == 1) { fy = y; break; }
  }
  float* pcol = posw + (size_t)b * HW + x;
  const float LOG2_DECAY = -0.32192809488736235f;   // log2(0.8)
  for (int y = 0; y < H; ++y) {
    int dist = fy - y;
    float w = 1.0f;
    if (dist > 0) w = 1.0f + 1.5f * exp2f((float)dist * LOG2_DECAY);
    pcol[y * W] = w;
  }
}

// ---------------------------------------------------------------------------
// Pass 4: fused log-softmax(C=3) CE + final_w + deterministic block partials.
// ---------------------------------------------------------------------------
__global__ __launch_bounds__(256)
void fused_kernel(const float* __restrict__ pred,
                  const int* __restrict__ target,
                  const int* __restrict__ mask,
                  const float* __restrict__ density,
                  const float* __restrict__ posw,
                  float* __restrict__ finalw,
                  float* __restrict__ partials) {
  __shared__ float sm[256];
  int p = blockIdx.x * 256 + threadIdx.x;
  float contrib = 0.0f;
  if (p < BHW) {
    int b  = p / HW;
    int yx = p - b * HW;
    const float* pp = pred + (size_t)b * NC * HW + yx;
    float p0 = pp[0], p1 = pp[HW], p2 = pp[2 * HW];
    float m = fmaxf(p0, fmaxf(p1, p2));
    float lse = m + __logf(__expf(p0 - m) + __expf(p1 - m) + __expf(p2 - m));
    int t = target[p];
    float pt = (t == 0) ? p0 : ((t == 1) ? p1 : p2);
    float ce = lse - pt;
    float fw = 1.0f + 2.0f * density[p] + 1.5f * posw[p];
    if (mask[p] == 1) fw = 1.0f;
    finalw[p] = fw;
    contrib = ce * fw;
  }
  sm[threadIdx.x] = contrib;
  __syncthreads();
  #pragma unroll
  for (int s = 128; s > 0; s >>= 1) {
    if (threadIdx.x < s) sm[threadIdx.x] += sm[threadIdx.x + s];
    __syncthreads();
  }
  if (threadIdx.x == 0) partials[blockIdx.x] = sm[0];
}

// Pass 5: fixed-order final reduction (deterministic, no float atomics).
__global__ __launch_bounds__(256)
void reduce_kernel(const float* __restrict__ partials, int n,
                   float* __restrict__ loss) {
  __shared__ float sm[256];
  float s = 0.0f;
  for (int i = threadIdx.x; i < n; i += 256) s += partials[i];
  sm[threadIdx.x] = s;
  __syncthreads();
  #pragma unroll
  for (int st = 128; st > 0; st >>= 1) {
    if (threadIdx.x < st) sm[threadIdx.x] += sm[threadIdx.x + st];
    __syncthreads();
  }
  if (threadIdx.x == 0) loss[0] = sm[0] / (float)BHW;
}

extern "C" void kernel_launch(void* const* d_in, const int* in_sizes, int n_in,
                              void* d_out, int out_size, void* d_ws, size_t ws_size,
                              hipStream_t stream) {
  (void)in_sizes; (void)n_in; (void)out_size; (void)ws_size;
  const float* pred   = (const float*)d_in[0];
  const int*   target = (const int*)d_in[1];
  const int*   mask   = (const int*)d_in[2];

  // d_out layout (return order): [loss][final_w BHW][density BHW][posw BHW]
  float* out     = (float*)d_out;
  float* loss    = out;
  float* finalw  = out + 1;
  float* density = out + 1 + (size_t)BHW;
  float* posw    = out + 1 + 2 * (size_t)BHW;

  // workspace: horizontal box-sum plane + block partials
  float* tmp      = (float*)d_ws;
  float* partials = tmp + (size_t)BHW;

  const int ntiles = BATCH * (H / 16) * (W / 16);   // 16384 tiles
  const int nblk   = BHW / 256;                      // 16384 blocks

  hbox_kernel<<<ntiles / 4, 128, 0, stream>>>(target, tmp);
  vbox_kernel<<<ntiles / 4, 128, 0, stream>>>(tmp, density);
  posw_kernel<<<(BATCH * W + 255) / 256, 256, 0, stream>>>(mask, posw);
  fused_kernel<<<nblk, 256, 0, stream>>>(pred, target, mask, density, posw,
                                         finalw, partials);
  reduce_kernel<<<1, 256, 0, stream>>>(partials, nblk, loss);
}